// AVIDMemory_75196287418425
// MI455X (gfx1250) — compile-verified
//
#include <hip/hip_runtime.h>
#include <hip/hip_bf16.h>
#include <math.h>

#define DIMS      128
#define BATCH     256
#define NEGS      1024
#define NSAMPLES  300000
#define SCORE_COLS 1025            // 1 positive + 1024 negatives

static constexpr float INV_TEMP = 1.0f / 0.07f;
static constexpr float F_EPS    = 1e-12f;

typedef __attribute__((ext_vector_type(2))) float v2f;
typedef __attribute__((ext_vector_type(8))) float v8f;

// ---------------------------------------------------------------------------
// Kernel 1: L2-normalize the batch embeddings (na = l2n(audio), nv = l2n(video))
// ---------------------------------------------------------------------------
__global__ __launch_bounds__(DIMS) void avid_norm_kernel(
    const float* __restrict__ ae, const float* __restrict__ ve,
    float* __restrict__ na, float* __restrict__ nv) {
  const int b = blockIdx.x;
  const int t = threadIdx.x;
  const float a = ae[b * DIMS + t];
  const float v = ve[b * DIMS + t];
  __shared__ float sa[DIMS];
  __shared__ float sv[DIMS];
  sa[t] = a * a;
  sv[t] = v * v;
  __syncthreads();
  for (int s = DIMS / 2; s > 0; s >>= 1) {
    if (t < s) { sa[t] += sa[t + s]; sv[t] += sv[t + s]; }
    __syncthreads();
  }
  const float da = fmaxf(sqrtf(sa[0]), F_EPS);
  const float dv = fmaxf(sqrtf(sv[0]), F_EPS);
  na[b * DIMS + t] = a / da;
  nv[b * DIMS + t] = v / dv;
}

// ---------------------------------------------------------------------------
// Kernel 2: negative scores via V_WMMA_F32_16X16X4_F32 (fp32-exact).
// One wave per (b, tile-of-16-negatives).
//   A (16x4 per step)  = 16 gathered memory rows (audio or video bank)
//   B (4x16 per step)  = col0 = na[b], col1 = nv[b], cols 2..15 = 0
//   D[m][0] = row_m . na[b],  D[m][1] = row_m . nv[b]
// Fragment layouts per CDNA5 ISA 7.12.2:
//   A: lanes 0-15 -> {K=4k,4k+1}, lanes 16-31 -> {K=4k+2,4k+3} (M = lane&15 row)
//   B: lanes 0-15 -> {K=4k,4k+1} of column N=lane, lanes 16-31 -> {K=4k+2,4k+3}
//   D: vgpr j: lanes 0-15 -> D[j][lane], lanes 16-31 -> D[j+8][lane-16]
// ---------------------------------------------------------------------------
__global__ __launch_bounds__(32) void avid_scores_wmma_kernel(
    const float* __restrict__ audio_mem, const float* __restrict__ video_mem,
    const float* __restrict__ na, const float* __restrict__ nv,
    const int* __restrict__ indices, const int* __restrict__ neg_samples,
    float* __restrict__ scores) {
  const int blk  = blockIdx.x;      // BATCH * (NEGS/16) blocks
  const int b    = blk >> 6;        // / 64 tiles
  const int tile = blk & 63;
  const int lane = threadIdx.x;     // wave32
  const int m    = lane & 15;       // matrix row (A) / column (B) index

  // alias-shifted negative index for row m of this tile (all lanes compute;
  // lanes 16-31 mirror lanes 0-15 -> no divergence, EXEC stays all-1s)
  const int pidx = indices[b];
  int ns = neg_samples[(size_t)b * NEGS + tile * 16 + m];
  const int nidx = ns + ((ns >= pidx) ? 1 : 0);

  const float* __restrict__ arow = audio_mem + (size_t)nidx * DIMS;
  const float* __restrict__ vrow = video_mem + (size_t)nidx * DIMS;
  const int koff = (lane < 16) ? 0 : 2;

  // B-matrix source: column 0 = na[b], column 1 = nv[b], others masked to 0
  const float* __restrict__ ctx = (m == 1) ? (nv + b * DIMS) : (na + b * DIMS);
  const float cmask = (m < 2) ? 1.0f : 0.0f;

  v8f accA = {};   // audio-bank rows x [na|nv]
  v8f accV = {};   // video-bank rows x [na|nv]
#pragma unroll 4
  for (int kk = 0; kk < DIMS / 4; ++kk) {
    const int kbase = kk * 4 + koff;            // 8-byte aligned
    v2f afrag = *(const v2f*)(arow + kbase);    // global_load_b64
    v2f vfrag = *(const v2f*)(vrow + kbase);
    v2f bfrag;
    bfrag.x = ctx[kbase + 0] * cmask;
    bfrag.y = ctx[kbase + 1] * cmask;
    accA = __builtin_amdgcn_wmma_f32_16x16x4_f32(
        false, afrag, false, bfrag, (short)0, accA, false, false);
    accV = __builtin_amdgcn_wmma_f32_16x16x4_f32(
        false, vfrag, false, bfrag, (short)0, accV, false, false);
  }

  // Scatter D columns 0 and 1 into the four score planes.
  const int col  = m;                       // N index held by this lane
  const int mofs = (lane < 16) ? 0 : 8;
  if (col < 2) {
    const size_t plane = (size_t)BATCH * SCORE_COLS;
    const size_t base  = (size_t)b * SCORE_COLS + 1 + (size_t)tile * 16 + mofs;
#pragma unroll
    for (int j = 0; j < 8; ++j) {
      const float sa = accA[j] * INV_TEMP;  // audio row . ctx[col]
      const float sv = accV[j] * INV_TEMP;  // video row . ctx[col]
      if (col == 0) {                       // ctx = na
        scores[2 * plane + base + j] = sa;  // intra audio : na . neg_a
        scores[0 * plane + base + j] = sv;  // inter a->v  : na . neg_v
      } else {                              // ctx = nv
        scores[1 * plane + base + j] = sa;  // inter v->a  : nv . neg_a
        scores[3 * plane + base + j] = sv;  // intra video : nv . neg_v
      }
    }
  }
}

// ---------------------------------------------------------------------------
// Kernel 3: positive scores + momentum memory row update (scatter over the
// already-copied output banks). One block per batch element.
// ---------------------------------------------------------------------------
__global__ __launch_bounds__(DIMS) void avid_pos_update_kernel(
    const float* __restrict__ amem, const float* __restrict__ vmem,
    const float* __restrict__ na, const float* __restrict__ nv,
    const int* __restrict__ indices, float* __restrict__ scores,
    float* __restrict__ aout, float* __restrict__ vout) {
  const int b = blockIdx.x;
  const int t = threadIdx.x;
  const int idx = indices[b];
  const float pa = amem[(size_t)idx * DIMS + t];
  const float pv = vmem[(size_t)idx * DIMS + t];
  const float xa = na[b * DIMS + t];
  const float xv = nv[b * DIMS + t];

  // four positive dot products: na.pv, nv.pa, na.pa, nv.pv  (score order 0..3)
  __shared__ float4 sm[DIMS];
  sm[t] = make_float4(xa * pv, xv * pa, xa * pa, xv * pv);
  __syncthreads();
  for (int s = DIMS / 2; s > 0; s >>= 1) {
    if (t < s) {
      sm[t].x += sm[t + s].x; sm[t].y += sm[t + s].y;
      sm[t].z += sm[t + s].z; sm[t].w += sm[t + s].w;
    }
    __syncthreads();
  }
  if (t < 4) {
    const float4 r = sm[0];
    const float dots[4] = {r.x, r.y, r.z, r.w};
    scores[(size_t)t * BATCH * SCORE_COLS + (size_t)b * SCORE_COLS] =
        dots[t] * INV_TEMP;
  }

  // momentum update: new = l2n(pos*0.5 + n*0.5), scattered at idx
  const float ua = pa * 0.5f + xa * 0.5f;
  const float uv = pv * 0.5f + xv * 0.5f;
  __shared__ float2 sm2[DIMS];
  sm2[t] = make_float2(ua * ua, uv * uv);
  __syncthreads();
  for (int s = DIMS / 2; s > 0; s >>= 1) {
    if (t < s) { sm2[t].x += sm2[t + s].x; sm2[t].y += sm2[t + s].y; }
    __syncthreads();
  }
  const float da = fmaxf(sqrtf(sm2[0].x), F_EPS);
  const float dv = fmaxf(sqrtf(sm2[0].y), F_EPS);
  aout[(size_t)idx * DIMS + t] = ua / da;
  vout[(size_t)idx * DIMS + t] = uv / dv;
}

// ---------------------------------------------------------------------------
extern "C" void kernel_launch(void* const* d_in, const int* in_sizes, int n_in,
                              void* d_out, int out_size, void* d_ws, size_t ws_size,
                              hipStream_t stream) {
  (void)in_sizes; (void)n_in; (void)out_size; (void)ws_size;

  const float* audio_emb = (const float*)d_in[0];   // [256,128]
  const float* video_emb = (const float*)d_in[1];   // [256,128]
  const float* amem      = (const float*)d_in[2];   // [300000,128]
  const float* vmem      = (const float*)d_in[3];   // [300000,128]
  const int*   indices   = (const int*)d_in[4];     // [256]
  const int*   negs      = (const int*)d_in[5];     // [256,1024]

  float* out    = (float*)d_out;
  float* scores = out;                                           // [4,256,1025]
  float* aout   = out + (size_t)4 * BATCH * SCORE_COLS;          // [300000,128]
  float* vout   = aout + (size_t)NSAMPLES * DIMS;                // [300000,128]

  float* na = (float*)d_ws;                  // [256,128]
  float* nv = na + BATCH * DIMS;             // [256,128]

  const size_t bank_bytes = (size_t)NSAMPLES * DIMS * sizeof(float);
  // Dominant bandwidth term: clone both memory banks into the output regions.
  hipMemcpyAsync(aout, amem, bank_bytes, hipMemcpyDeviceToDevice, stream);
  hipMemcpyAsync(vout, vmem, bank_bytes, hipMemcpyDeviceToDevice, stream);

  avid_norm_kernel<<<BATCH, DIMS, 0, stream>>>(audio_emb, video_emb, na, nv);

  avid_scores_wmma_kernel<<<BATCH * (NEGS / 16), 32, 0, stream>>>(
      amem, vmem, na, nv, indices, negs, scores);

  avid_pos_update_kernel<<<BATCH, DIMS, 0, stream>>>(
      amem, vmem, na, nv, indices, scores, aout, vout);
}